// ANFIS_18073222381745
// MI455X (gfx1250) — compile-verified
//
#include <hip/hip_runtime.h>
#include <math.h>

// ANFIS forward for MI455X / gfx1250 (wave32, WMMA, LDS-staged).
// d_out layout (fp32): out[B] | norm_firing[B,128] | mem[B,7,2]
#define N_FEAT  7
#define N_RULES 128
#define BATCH   65536
#define BLOCK   64            // 2 wave32 per block
#define FSTRIDE 132           // padded LDS row stride (16B aligned, bank-friendly)

typedef float v2f __attribute__((ext_vector_type(2)));
typedef float v8f __attribute__((ext_vector_type(8)));

__device__ __forceinline__ float sigmoidf_(float v) { return 1.0f / (1.0f + expf(-v)); }

__global__ __launch_bounds__(BLOCK)
void anfis_fwd_wmma(const float* __restrict__ x,       // [B,7]
                    const float* __restrict__ mu,      // [7,2]
                    const float* __restrict__ sigma,   // [7,2]
                    const float* __restrict__ ta,
                    const float* __restrict__ tb,
                    const float* __restrict__ tc,
                    const float* __restrict__ td,
                    const float* __restrict__ mf_mix,  // [1]
                    const float* __restrict__ tnw,     // [1]
                    const float* __restrict__ cq,      // [128,8]
                    const int*   __restrict__ ridx,    // [128,7]
                    float* __restrict__ out,           // [B]
                    float* __restrict__ nf_out,        // [B,128]
                    float* __restrict__ mem_out)       // [B,14]
{
    __shared__ __align__(16) float s_fir[BLOCK][FSTRIDE]; // raw firings
    __shared__ float s_inv[BLOCK];                        // per-row 1/(sum+eps)
    __shared__ float s_mem[BLOCK][N_FEAT * 2];
    __shared__ int   s_ridx[N_RULES * N_FEAT];

    const int tid = threadIdx.x;
    for (int i = tid; i < N_RULES * N_FEAT; i += BLOCK) s_ridx[i] = ridx[i];
    __syncthreads();

    const int   row   = blockIdx.x * BLOCK + tid;
    const float alpha = sigmoidf_(mf_mix[0]);
    const float w     = sigmoidf_(tnw[0]);

    // ---------- Phase 1: memberships + raw firings (one row per lane) ------
    float xv[N_FEAT], m0[N_FEAT], m1[N_FEAT];
#pragma unroll
    for (int f = 0; f < N_FEAT; ++f) xv[f] = x[row * N_FEAT + f];

#pragma unroll
    for (int f = 0; f < N_FEAT; ++f) {
#pragma unroll
        for (int s = 0; s < 2; ++s) {
            const int   p  = f * 2 + s;
            const float xf = xv[f];
            const float sg = fmaxf(sigma[p], 1e-6f);
            const float dm = xf - mu[p];
            const float gauss = expf(-(dm * dm) / (2.0f * sg * sg));
            float pa = ta[p], pb = tb[p], pc = tc[p], pd = td[p], t;
            if (pa > pb) { t = pa; pa = pb; pb = t; }
            if (pc > pd) { t = pc; pc = pd; pd = t; }
            if (pa > pc) { t = pa; pa = pc; pc = t; }
            if (pb > pd) { t = pb; pb = pd; pd = t; }
            if (pb > pc) { t = pb; pb = pc; pc = t; }
            const float left  = fminf(fmaxf((xf - pa) / (pb - pa + 1e-6f), 0.f), 1.f);
            const float flat  = (xf >= pb && xf <= pc) ? 1.f : 0.f;
            const float right = fminf(fmaxf((pd - xf) / (pd - pc + 1e-6f), 0.f), 1.f);
            const float trap  = fmaxf(fminf(left, right), flat);
            const float mv    = alpha * gauss + (1.f - alpha) * trap;
            if (s == 0) m0[f] = mv; else m1[f] = mv;
            s_mem[tid][p] = mv;
        }
    }

    float fsum = 0.f;
    for (int r = 0; r < N_RULES; ++r) {
        float pr = 1.f, mn = 3.402823466e38f;
#pragma unroll
        for (int f = 0; f < N_FEAT; ++f) {
            const int   idx = s_ridx[r * N_FEAT + f];
            const float v   = idx ? m1[f] : m0[f];
            pr *= v;
            mn = fminf(mn, v);
        }
        const float fir = w * pr + (1.f - w) * mn;
        fsum += fir;
        s_fir[tid][r] = fir;
    }
    s_inv[tid] = 1.f / (fsum + 1e-8f);

    __syncthreads();

    // ---------- Coalesced output copies ------------------------------------
    // mem: 64*14 = 896 floats per block, block-cooperative, fully coalesced.
    {
        const int base = blockIdx.x * BLOCK * (N_FEAT * 2);
        for (int k = tid; k < BLOCK * N_FEAT * 2; k += BLOCK)
            mem_out[base + k] = s_mem[k / (N_FEAT * 2)][k % (N_FEAT * 2)];
    }

    const int wv   = tid >> 5;
    const int lane = tid & 31;
    const int half = lane >> 4;   // 0: lanes 0-15, 1: lanes 16-31
    const int n    = lane & 15;

    // norm_firing: per wave, one row per iteration, float4 stores (512B/instr)
    {
        for (int rr = 0; rr < 32; ++rr) {
            const int   rl  = wv * 32 + rr;
            const int   g   = blockIdx.x * BLOCK + rl;
            const float inv = s_inv[rl];
            float4 v = *(const float4*)&s_fir[rl][lane * 4];
            v.x *= inv; v.y *= inv; v.z *= inv; v.w *= inv;
            *(float4*)&nf_out[(size_t)g * N_RULES + lane * 4] = v;
        }
    }

    // ---------- Phase 2: rule_out via V_WMMA_F32_16X16X4_F32 ---------------
    // rule_out(16x128) = x_aug(16x8) @ consequent^T(8x128); K=8 = 2 ksteps.
    // A 16x4 f32: lanes 0-15: M=lane, V0=K0 V1=K1; lanes 16-31: M=lane-16, V0=K2 V1=K3.
    // B 4x16 f32 (mirrored): lanes 0-15 V0=K0,V1=K1 rows; lanes 16-31 V0=K2,V1=K3; N=lane&15.
#pragma unroll
    for (int hg = 0; hg < 2; ++hg) {
        const int rl_base = wv * 32 + hg * 16;            // local row base
        const int rbase   = blockIdx.x * BLOCK + rl_base; // global row base
        const int arow    = rbase + n;
        v2f a0, a1;
        if (half == 0) {
            a0[0] = 1.0f;            a0[1] = x[arow * 7 + 0];  // K=0,1
            a1[0] = x[arow * 7 + 3]; a1[1] = x[arow * 7 + 4];  // K=4,5
        } else {
            a0[0] = x[arow * 7 + 1]; a0[1] = x[arow * 7 + 2];  // K=2,3
            a1[0] = x[arow * 7 + 5]; a1[1] = x[arow * 7 + 6];  // K=6,7
        }

        v8f acc;
#pragma unroll
        for (int i = 0; i < 8; ++i) acc[i] = 0.f;

        for (int tile = 0; tile < 8; ++tile) {
            const int    rule = tile * 16 + n;
            const float* c0   = cq + rule * 8;
            v2f b0, b1;
            b0[0] = c0[2 * half];     b0[1] = c0[2 * half + 1];
            b1[0] = c0[4 + 2 * half]; b1[1] = c0[4 + 2 * half + 1];

            v8f cz;
#pragma unroll
            for (int i = 0; i < 8; ++i) cz[i] = 0.f;

            v8f d = __builtin_amdgcn_wmma_f32_16x16x4_f32(
                        false, a0, false, b0, (short)0, cz, false, false);
            d = __builtin_amdgcn_wmma_f32_16x16x4_f32(
                        false, a1, false, b1, (short)0, d, false, false);

            // weight by raw firing from LDS; D VGPR i = row i (lanes 0-15) /
            // row i+8 (lanes 16-31), N = lane&15.
#pragma unroll
            for (int i = 0; i < 8; ++i)
                acc[i] += d[i] * s_fir[rl_base + i + 8 * half][rule];
        }

        // reduce the 16 lanes of each half (rule-column dimension)
#pragma unroll
        for (int i = 0; i < 8; ++i) {
            float v = acc[i];
            v += __shfl_xor(v, 1, 32);
            v += __shfl_xor(v, 2, 32);
            v += __shfl_xor(v, 4, 32);
            v += __shfl_xor(v, 8, 32);
            acc[i] = v;
        }
        if (n == 0) {
#pragma unroll
            for (int i = 0; i < 8; ++i) {
                const int rl = rl_base + i + 8 * half;
                out[blockIdx.x * BLOCK + rl] = acc[i] * s_inv[rl];
            }
        }
    }
}

extern "C" void kernel_launch(void* const* d_in, const int* in_sizes, int n_in,
                              void* d_out, int out_size, void* d_ws, size_t ws_size,
                              hipStream_t stream) {
    (void)in_sizes; (void)n_in; (void)d_ws; (void)ws_size; (void)out_size;
    const float* x     = (const float*)d_in[0];
    const float* mu    = (const float*)d_in[1];
    const float* sigma = (const float*)d_in[2];
    const float* ta    = (const float*)d_in[3];
    const float* tb    = (const float*)d_in[4];
    const float* tc    = (const float*)d_in[5];
    const float* td    = (const float*)d_in[6];
    const float* mfm   = (const float*)d_in[7];
    const float* tnw   = (const float*)d_in[8];
    const float* cq    = (const float*)d_in[9];
    const int*   ridx  = (const int*)d_in[10];

    float* out     = (float*)d_out;
    float* nf_out  = out + BATCH;
    float* mem_out = out + BATCH + (size_t)BATCH * N_RULES;

    anfis_fwd_wmma<<<BATCH / BLOCK, BLOCK, 0, stream>>>(
        x, mu, sigma, ta, tb, tc, td, mfm, tnw, cq, ridx,
        out, nf_out, mem_out);
}